// ChamferLoss_11948599017824
// MI455X (gfx1250) — compile-verified
//
#include <hip/hip_runtime.h>

typedef __attribute__((ext_vector_type(2))) float v2f;
typedef __attribute__((ext_vector_type(8))) float v8f;

#define BATCH 4
#define NPTS  8192
#define CTILES (NPTS / 16)                 // 512 column tiles per batch
#define ROW_TILES (BATCH * CTILES)         // 2048 row tiles total
#define WAVES_PER_BLOCK 8
#define THREADS (WAVES_PER_BLOCK * 32)

// ---------------------------------------------------------------------------
// Pre-pass: pack per-16-col WMMA B fragments for one point array P [B,3,N].
// Fragment tile = 32 lanes x v2f:
//   lanes 0-15:  (p0, p1)      -> K0, K1 rows of B
//   lanes 16-31: (p2, |p|^2)   -> K2, K3 rows of B   (y^2 folded into K=3)
// frag layout: [B * CTILES * 32] v2f, index = ((b*CTILES + t)*32 + lane)
// ---------------------------------------------------------------------------
__global__ __launch_bounds__(256) void pack_frag_kernel(
    const float* __restrict__ P, v2f* __restrict__ frag)
{
    const int gid  = blockIdx.x * blockDim.x + threadIdx.x;  // B*CTILES*32 total
    const int lane = gid & 31;
    const int tb   = gid >> 5;                 // b*CTILES + t
    const int b    = tb >> 9;
    const int col  = ((tb & 511) << 4) + (lane & 15);
    const bool hi  = lane >= 16;

    const float* __restrict__ base = P + b * 3 * NPTS;
    float p0 = base[col];
    float p1 = base[NPTS + col];
    float p2 = base[2 * NPTS + col];
    float pp = fmaf(p0, p0, fmaf(p1, p1, p2 * p2));

    v2f f;
    f.x = hi ? p2 : p0;
    f.y = hi ? pp : p1;
    frag[gid] = f;
}

// ---------------------------------------------------------------------------
// One wave32 computes dist[b, n0..n0+15] = min over all m of max(0,||x_n-y_m||^2)
// via V_WMMA_F32_16X16X4_F32:
//   A = [-2x0, -2x1, -2x2, 1] (16x4, from packed row frag)
//   B = [y0; y1; y2; y^2]     (4x16, one b64 load from packed col frag)
//   D = A*B (+0 inline)  = y^2 - 2 x.y
// Hot loop: 1 global_load_b64 + 1 WMMA + 8 min per 16x16 tile.
// ---------------------------------------------------------------------------
__global__ __launch_bounds__(THREADS) void chamfer_rowmin_kernel(
    const v2f* __restrict__ fragRow,   // packed frags of the row array
    const v2f* __restrict__ fragCol,   // packed frags of the col array
    const float* __restrict__ Prow,    // original row array [B,3,N] (epilogue x^2)
    float* __restrict__ dist)          // [B,N]
{
    const int lane = threadIdx.x & 31;
    const int wave = (blockIdx.x * blockDim.x + threadIdx.x) >> 5;  // b*CTILES + rowtile
    const int b    = wave >> 9;
    const int n0   = (wave & 511) << 4;

    const bool hi  = lane >= 16;

    // ---- A fragment from packed row frags: a = (-2*fx, hi ? 1 : -2*fy) ----
    v2f fa = fragRow[(wave << 5) + lane];
    v2f a;
    a.x = -2.0f * fa.x;
    a.y = hi ? 1.0f : (-2.0f * fa.y);

    float rmin[8];
#pragma unroll
    for (int i = 0; i < 8; ++i) rmin[i] = 3.0e38f;

    const v2f* __restrict__ fb = fragCol + (b << 14) + lane;  // b*CTILES*32

#pragma unroll 8
    for (int t = 0; t < CTILES; ++t) {
        v2f bf = fb[t << 5];              // stride 32 v2f = 256B per tile

        v8f czero = {};
        v8f d = __builtin_amdgcn_wmma_f32_16x16x4_f32(
            /*neg_a=*/false, a, /*neg_b=*/false, bf,
            /*c_mod=*/(short)0, czero, /*reuse_a=*/false, /*reuse_b=*/false);

#pragma unroll
        for (int i = 0; i < 8; ++i) rmin[i] = fminf(rmin[i], d[i]);
    }

    // ---- cross-lane min within each 16-lane half (lanes are columns) ----
#pragma unroll
    for (int off = 8; off >= 1; off >>= 1) {
#pragma unroll
        for (int i = 0; i < 8; ++i)
            rmin[i] = fminf(rmin[i], __shfl_xor(rmin[i], off, 32));
    }

    // Lane 0 holds row-mins for rows r (VGPR r), lane 16 for rows r+8.
    if (lane == 0 || lane == 16) {
        const int rbase = hi ? 8 : 0;
        const float* __restrict__ xb = Prow + b * 3 * NPTS;
#pragma unroll
        for (int r = 0; r < 8; ++r) {
            const int row = n0 + rbase + r;
            float x0 = xb[row];
            float x1 = xb[NPTS + row];
            float x2 = xb[2 * NPTS + row];
            float xx = x0 * x0 + x1 * x1 + x2 * x2;
            dist[b * NPTS + row] = fmaxf(0.0f, rmin[r] + xx);
        }
    }
}

// mean(dx) + 10*mean(dy)  -> single scalar
__global__ __launch_bounds__(256) void chamfer_reduce_kernel(
    const float* __restrict__ dx, const float* __restrict__ dy,
    float* __restrict__ out)
{
    __shared__ float s[256];
    const int n = BATCH * NPTS;
    float acc = 0.0f;
    for (int i = threadIdx.x; i < n; i += 256)
        acc += dx[i] + 10.0f * dy[i];
    s[threadIdx.x] = acc;
    __syncthreads();
#pragma unroll
    for (int off = 128; off > 0; off >>= 1) {
        if (threadIdx.x < off) s[threadIdx.x] += s[threadIdx.x + off];
        __syncthreads();
    }
    if (threadIdx.x == 0) out[0] = s[0] / (float)n;
}

extern "C" void kernel_launch(void* const* d_in, const int* in_sizes, int n_in,
                              void* d_out, int out_size, void* d_ws, size_t ws_size,
                              hipStream_t stream) {
    const float* x = (const float*)d_in[0];   // [4,3,8192]
    const float* y = (const float*)d_in[1];   // [4,3,8192]
    float* out = (float*)d_out;

    float* wsf    = (float*)d_ws;
    float* dist_x = wsf;                              // [B*N]
    float* dist_y = dist_x + BATCH * NPTS;            // [B*N]
    v2f*   fragX  = (v2f*)(dist_y + BATCH * NPTS);    // [B*CTILES*32]
    v2f*   fragY  = fragX + BATCH * CTILES * 32;      // [B*CTILES*32]

    // Pack WMMA fragments once per array (B*CTILES*32 = 65536 threads each).
    dim3 pgrid(BATCH * CTILES * 32 / 256), pblock(256);
    pack_frag_kernel<<<pgrid, pblock, 0, stream>>>(x, fragX);
    pack_frag_kernel<<<pgrid, pblock, 0, stream>>>(y, fragY);

    dim3 grid(ROW_TILES / WAVES_PER_BLOCK);           // 256 blocks
    dim3 block(THREADS);                              // 256 threads = 8 waves

    // Pass 1: for each x_n, min over y_m
    chamfer_rowmin_kernel<<<grid, block, 0, stream>>>(fragX, fragY, x, dist_x);
    // Pass 2: for each y_m, min over x_n (symmetric)
    chamfer_rowmin_kernel<<<grid, block, 0, stream>>>(fragY, fragX, y, dist_y);

    chamfer_reduce_kernel<<<1, 256, 0, stream>>>(dist_x, dist_y, out);
}